// DetectionPostProcessor_58171037057714
// MI455X (gfx1250) — compile-verified
//
#include <hip/hip_runtime.h>
#include <stdint.h>

// ---------------- problem constants (from reference) ----------------
#define B_DIM 4
#define C_DIM 10
#define H_DIM 512
#define W_DIM 512
#define HW_DIM (H_DIM * W_DIM)            // 262144
#define CHW_DIM (C_DIM * HW_DIM)          // 2621440
#define N_ELEMS (B_DIM * CHW_DIM)         // 10485760
#define N4 (N_ELEMS / 4)                  // 2621440 float4s
#define K_PRE 2048
#define CAND_CAP 4096
#define MAX_DET 100
#define SCORE_THR 0.3f
#define NMS_THR 0.5f

typedef unsigned int uint32;
typedef unsigned long long u64;
typedef __attribute__((ext_vector_type(2))) float v2f;
typedef __attribute__((ext_vector_type(8))) float v8f;
typedef __attribute__((ext_vector_type(4))) int v4i;

// ---------------- CDNA5 feature guards ----------------
#if defined(__gfx1250__) && __has_builtin(__builtin_amdgcn_global_load_async_to_lds_b128)
#define HAVE_ASYNC_LDS 1
#else
#define HAVE_ASYNC_LDS 0
#endif

#if defined(__gfx1250__) && __has_builtin(__builtin_amdgcn_wmma_f32_16x16x4_f32)
#define HAVE_WMMA_F32X4 1
#else
#define HAVE_WMMA_F32X4 0
#endif

// AS-qualified pointer types for the async builtin (global = AS1, LDS = AS3).
typedef __attribute__((address_space(1))) v4i* gas_v4i_ptr;
typedef __attribute__((address_space(3))) v4i* las_v4i_ptr;

static __device__ __forceinline__ void asyncCopy16(const float4* g, float4* lds) {
#if HAVE_ASYNC_LDS
  __builtin_amdgcn_global_load_async_to_lds_b128(
      (gas_v4i_ptr)(uintptr_t)g,
      (las_v4i_ptr)(uint32)(uintptr_t)lds,
      0, 0);
#else
  *lds = *g;
#endif
}

static __device__ __forceinline__ void asyncWait0() {
#if HAVE_ASYNC_LDS
#if __has_builtin(__builtin_amdgcn_s_wait_asynccnt)
  __builtin_amdgcn_s_wait_asynccnt(0);
#else
  asm volatile("s_wait_asynccnt 0" ::: "memory");
#endif
#endif
}

static __device__ __forceinline__ unsigned ballot32(bool p) {
#if __has_builtin(__builtin_amdgcn_ballot_w32)
  return __builtin_amdgcn_ballot_w32(p);
#else
  return (unsigned)__ballot(p);
#endif
}

// monotonic order-preserving float->uint key
static __device__ __forceinline__ uint32 fkey(float f) {
  uint32 b = __float_as_uint(f);
  return b ^ ((b & 0x80000000u) ? 0xFFFFFFFFu : 0x80000000u);
}
static __device__ __forceinline__ float keyToFloat(uint32 k) {
  uint32 b = (k & 0x80000000u) ? (k ^ 0x80000000u) : ~k;
  return __uint_as_float(b);
}

// ---------------- workspace layout ----------------
struct Ws {
  uint32* hist1;        // 65536
  uint32* hist2;        // 65536
  uint32* scal;         // 64: [0]=cand counter [1]=cut_hi [2]=need [3]=threshold key
  uint32* keepw;        // 64 keep-mask words
  u64*    cand;         // CAND_CAP
  u64*    sorted;       // K_PRE
  float*  boxes;        // K_PRE*7
  float*  lox;          // K_PRE
  float*  loy;          // K_PRE
  float*  hix;          // K_PRE
  float*  hiy;          // K_PRE
  float*  area;         // K_PRE
  float*  score;        // K_PRE
  int*    label;        // K_PRE
  uint32* validf;       // K_PRE
  unsigned short* sup16; // K_PRE * 128 halfwords
};

static __host__ __device__ __forceinline__ Ws makeWs(void* base) {
  Ws w;
  char* p = (char*)base;
  w.hist1  = (uint32*)p;                 p += 65536 * 4;
  w.hist2  = (uint32*)p;                 p += 65536 * 4;
  w.scal   = (uint32*)p;                 p += 64 * 4;
  w.keepw  = (uint32*)p;                 p += 64 * 4;
  w.cand   = (u64*)p;                    p += CAND_CAP * 8;
  w.sorted = (u64*)p;                    p += K_PRE * 8;
  w.boxes  = (float*)p;                  p += K_PRE * 7 * 4;
  w.lox    = (float*)p;                  p += K_PRE * 4;
  w.loy    = (float*)p;                  p += K_PRE * 4;
  w.hix    = (float*)p;                  p += K_PRE * 4;
  w.hiy    = (float*)p;                  p += K_PRE * 4;
  w.area   = (float*)p;                  p += K_PRE * 4;
  w.score  = (float*)p;                  p += K_PRE * 4;
  w.label  = (int*)p;                    p += K_PRE * 4;
  w.validf = (uint32*)p;                 p += K_PRE * 4;
  w.sup16  = (unsigned short*)p;         p += K_PRE * 128 * 2;
  return w;
}

// ---------------- kernels ----------------
__global__ void dpp_zero(uint32* p, int n) {
  int i = blockIdx.x * blockDim.x + threadIdx.x;
  if (i < n) p[i] = 0;
}

// Pass 1: histogram of key>>16, streamed via async global->LDS double buffering.
__global__ void dpp_hist_hi(const float* __restrict__ scores, void* wsbase) {
  Ws ws = makeWs(wsbase);
  __shared__ float4 buf[2][256];
  const int tid = threadIdx.x;
  const int TPB_TILES = 8;  // 1280 blocks * 8 tiles * 256 f4 = N4
  long tile0 = (long)blockIdx.x * TPB_TILES;
  const float4* s4 = (const float4*)scores;

  asyncCopy16(&s4[tile0 * 256 + tid], &buf[0][tid]);
  for (int t = 0; t < TPB_TILES; ++t) {
    asyncWait0();
    __syncthreads();
    int cur = t & 1;
    if (t + 1 < TPB_TILES)
      asyncCopy16(&s4[(tile0 + t + 1) * 256 + tid], &buf[(t + 1) & 1][tid]);
    float4 v = buf[cur][tid];
    atomicAdd(&ws.hist1[fkey(v.x) >> 16], 1u);
    atomicAdd(&ws.hist1[fkey(v.y) >> 16], 1u);
    atomicAdd(&ws.hist1[fkey(v.z) >> 16], 1u);
    atomicAdd(&ws.hist1[fkey(v.w) >> 16], 1u);
    __syncthreads();
  }
}

// Scan a 65536-bin histogram from high to low to find the K-th cut bin.
__global__ void dpp_scan(void* wsbase, int stage) {
  Ws ws = makeWs(wsbase);
  const uint32* hist = (stage == 0) ? ws.hist1 : ws.hist2;
  __shared__ uint32 part[1024];
  int t = threadIdx.x;
  uint32 s = 0;
  for (int b = 0; b < 64; ++b) s += hist[t * 64 + b];
  part[t] = s;
  __syncthreads();
  if (t == 0) {
    uint32 target = (stage == 0) ? (uint32)K_PRE : ws.scal[2];
    uint32 acc = 0, need = 0;
    int cut = 0;
    for (int seg = 1023; seg >= 0; --seg) {
      if (acc + part[seg] >= target) {
        for (int b = 63; b >= 0; --b) {
          int bin = seg * 64 + b;
          uint32 cv = hist[bin];
          if (acc + cv >= target) { cut = bin; need = target - acc; break; }
          acc += cv;
        }
        break;
      }
      acc += part[seg];
    }
    if (stage == 0) { ws.scal[1] = (uint32)cut; ws.scal[2] = need; }
    else            { ws.scal[3] = (ws.scal[1] << 16) | (uint32)cut; }
  }
}

// Pass 2: histogram of low 16 key bits for elements whose high bits == cut_hi.
__global__ void dpp_hist_lo(const float* __restrict__ scores, void* wsbase) {
  Ws ws = makeWs(wsbase);
  uint32 cutHi = ws.scal[1];
  const float4* s4 = (const float4*)scores;
  int stride = gridDim.x * blockDim.x;
  for (int i = blockIdx.x * blockDim.x + threadIdx.x; i < N4; i += stride) {
    __builtin_prefetch(&s4[i + stride], 0, 0);
    float4 v = s4[i];
    uint32 k0 = fkey(v.x), k1 = fkey(v.y), k2 = fkey(v.z), k3 = fkey(v.w);
    if ((k0 >> 16) == cutHi) atomicAdd(&ws.hist2[k0 & 0xFFFFu], 1u);
    if ((k1 >> 16) == cutHi) atomicAdd(&ws.hist2[k1 & 0xFFFFu], 1u);
    if ((k2 >> 16) == cutHi) atomicAdd(&ws.hist2[k2 & 0xFFFFu], 1u);
    if ((k3 >> 16) == cutHi) atomicAdd(&ws.hist2[k3 & 0xFFFFu], 1u);
  }
}

// Pass 3: compact all elements with key >= threshold.
__global__ void dpp_compact(const float* __restrict__ scores, void* wsbase) {
  Ws ws = makeWs(wsbase);
  uint32 thr = ws.scal[3];
  const float4* s4 = (const float4*)scores;
  int stride = gridDim.x * blockDim.x;
  for (int i = blockIdx.x * blockDim.x + threadIdx.x; i < N4; i += stride) {
    __builtin_prefetch(&s4[i + stride], 0, 0);
    float4 v = s4[i];
    float f[4] = {v.x, v.y, v.z, v.w};
#pragma unroll
    for (int c = 0; c < 4; ++c) {
      uint32 k = fkey(f[c]);
      if (k >= thr) {
        uint32 pos = atomicAdd(&ws.scal[0], 1u);
        if (pos < CAND_CAP) {
          uint32 idx = (uint32)(4 * i + c);
          ws.cand[pos] = ((u64)k << 32) | (u64)(~idx);  // (score desc, idx asc)
        }
      }
    }
  }
}

// Single-WGP bitonic sort of <=4096 candidates; emit top K_PRE descending.
__global__ void dpp_sort(void* wsbase) {
  Ws ws = makeWs(wsbase);
  __shared__ u64 keys[CAND_CAP];
  int tid = threadIdx.x;
  uint32 cnt = ws.scal[0];
  if (cnt > CAND_CAP) cnt = CAND_CAP;
  for (int i = tid; i < CAND_CAP; i += 1024)
    keys[i] = (i < (int)cnt) ? ws.cand[i] : 0ull;
  __syncthreads();
  for (int k = 2; k <= CAND_CAP; k <<= 1) {
    for (int j = k >> 1; j > 0; j >>= 1) {
      for (int i = tid; i < CAND_CAP; i += 1024) {
        int ixj = i ^ j;
        if (ixj > i) {
          bool up = ((i & k) == 0);  // ascending bitonic
          u64 a = keys[i], b = keys[ixj];
          if ((a > b) == up) { keys[i] = b; keys[ixj] = a; }
        }
      }
      __syncthreads();
    }
  }
  for (int i = tid; i < K_PRE; i += 1024)
    ws.sorted[i] = keys[CAND_CAP - 1 - i];  // descending
}

// Decode boxes for the top-2048 candidates.
__global__ void dpp_decode(const float* __restrict__ bbox, void* wsbase) {
  Ws ws = makeWs(wsbase);
  int t = blockIdx.x * blockDim.x + threadIdx.x;
  if (t >= K_PRE) return;
  u64 comp = ws.sorted[t];
  uint32 k = (uint32)(comp >> 32);
  uint32 idx = ~(uint32)(comp & 0xFFFFFFFFu);
  float sc = keyToFloat(k);
  int b = (int)(idx / CHW_DIM);
  int rem = (int)(idx % CHW_DIM);
  int cls = rem / HW_DIM;
  int rem2 = rem % HW_DIM;
  int h = rem2 / W_DIM;
  int w = rem2 % W_DIM;
  const float* bp = bbox + (size_t)b * 7 * HW_DIM + (size_t)h * W_DIM + w;
  float p0 = bp[0 * HW_DIM], p1 = bp[1 * HW_DIM], p2 = bp[2 * HW_DIM];
  float p3 = bp[3 * HW_DIM], p4 = bp[4 * HW_DIM], p5 = bp[5 * HW_DIM];
  float p6 = bp[6 * HW_DIM];
  float x = -51.2f + ((float)w + 0.5f) * 0.2f + p0;
  float y = -51.2f + ((float)h + 0.5f) * 0.2f + p1;
  float bw = expf(p3);
  float bl = expf(p4);
  float bh = expf(p5);
  ws.boxes[t * 7 + 0] = x;  ws.boxes[t * 7 + 1] = y;  ws.boxes[t * 7 + 2] = p2;
  ws.boxes[t * 7 + 3] = bw; ws.boxes[t * 7 + 4] = bl; ws.boxes[t * 7 + 5] = bh;
  ws.boxes[t * 7 + 6] = p6;
  float hw = bw * 0.5f, hl = bl * 0.5f;
  ws.lox[t] = x - hw; ws.loy[t] = y - hl;
  ws.hix[t] = x + hw; ws.hiy[t] = y + hl;
  ws.area[t] = bw * bl;
  ws.score[t] = sc;
  ws.label[t] = cls;
  ws.validf[t] = (sc > SCORE_THR) ? 1u : 0u;
}

// Suppression matrix: one wave per 16x16 tile. Union base (area_i + area_j)
// comes from v_wmma_f32_16x16x4_f32 (outer-sum via rank-2 matmul).
__global__ void dpp_suppress(void* wsbase) {
  Ws ws = makeWs(wsbase);
  int wave = (int)((blockIdx.x * blockDim.x + threadIdx.x) >> 5);
  int lane = (int)(threadIdx.x & 31);
  int tr = wave >> 7;      // 128 row tiles
  int tc = wave & 127;     // 128 col tiles
  int nlo = lane & 15;
  bool hiHalf = lane >= 16;
  int rbase = tr * 16 + (hiHalf ? 8 : 0);

  float ubase[8];
#if HAVE_WMMA_F32X4
  {
    float arow = ws.area[tr * 16 + nlo];
    float acol = ws.area[tc * 16 + nlo];
    v2f a, b;
    a.x = hiHalf ? 0.0f : arow;  a.y = hiHalf ? 0.0f : 1.0f;   // A: K0=area_i, K1=1
    b.x = hiHalf ? 0.0f : 1.0f;  b.y = hiHalf ? 0.0f : acol;   // B: K0=1, K1=area_j
    v8f c = {0.f, 0.f, 0.f, 0.f, 0.f, 0.f, 0.f, 0.f};
    c = __builtin_amdgcn_wmma_f32_16x16x4_f32(false, a, false, b, (short)0, c,
                                              false, false);
#pragma unroll
    for (int r = 0; r < 8; ++r) ubase[r] = c[r];
  }
#else
  {
    float acol = ws.area[tc * 16 + nlo];
#pragma unroll
    for (int r = 0; r < 8; ++r) ubase[r] = ws.area[rbase + r] + acol;
  }
#endif

  int j = tc * 16 + nlo;
  float jlx = ws.lox[j], jly = ws.loy[j], jhx = ws.hix[j], jhy = ws.hiy[j];
  int jl = ws.label[j];
  bool jv = ws.validf[j] != 0u;

#pragma unroll
  for (int r = 0; r < 8; ++r) {
    int i = rbase + r;
    float iw = fminf(ws.hix[i], jhx) - fmaxf(ws.lox[i], jlx);
    float ih = fminf(ws.hiy[i], jhy) - fmaxf(ws.loy[i], jly);
    iw = fmaxf(iw, 0.0f);
    ih = fmaxf(ih, 0.0f);
    float inter = iw * ih;
    float uni = ubase[r] - inter;
    float iou = inter / (uni + 1e-6f);
    bool sup = (iou >= NMS_THR) && (ws.label[i] == jl) && (j > i) && jv;
    unsigned m = ballot32(sup);
    if (lane == 0) {
      ws.sup16[(tr * 16 + r) * 128 + tc] = (unsigned short)(m & 0xFFFFu);
      ws.sup16[(tr * 16 + r + 8) * 128 + tc] = (unsigned short)(m >> 16);
    }
  }
}

// Greedy NMS recurrence on one wave32: keep mask register-distributed,
// suppress rows software-pipelined (unconditional lookahead loads).
__global__ void dpp_nms(void* wsbase) {
  Ws ws = makeWs(wsbase);
  const uint32* sup32 = (const uint32*)ws.sup16;  // [2048][64]
  int L = (int)threadIdx.x;                       // 0..31
  uint32 k0 = 0, k1 = 0;
  for (int b = 0; b < 32; ++b) {
    k0 |= (ws.validf[L * 64 + b] & 1u) << b;
    k1 |= (ws.validf[L * 64 + 32 + b] & 1u) << b;
  }
  uint32 s0 = sup32[0 * 64 + 2 * L];
  uint32 s1 = sup32[0 * 64 + 2 * L + 1];
  for (int i = 0; i < K_PRE; ++i) {
    uint32 ns0 = 0, ns1 = 0;
    if (i + 1 < K_PRE) {
      ns0 = sup32[(i + 1) * 64 + 2 * L];
      ns1 = sup32[(i + 1) * 64 + 2 * L + 1];
    }
    int owner = i >> 6;
    int sel = (i >> 5) & 1;
    int bit = i & 31;
    uint32 w0 = (uint32)__shfl((int)k0, owner, 32);
    uint32 w1 = (uint32)__shfl((int)k1, owner, 32);
    uint32 wv = sel ? w1 : w0;
    if ((wv >> bit) & 1u) {   // keep[i] set -> suppress its victims
      k0 &= ~s0;
      k1 &= ~s1;
    }
    s0 = ns0;
    s1 = ns1;
  }
  ws.keepw[2 * L] = k0;
  ws.keepw[2 * L + 1] = k1;
}

// Final top-100 emit. Array is already sorted descending, so kept entries in
// index order == top_k(kept_scores); BIG_NEG fillers are lowest-index unkept.
__global__ void dpp_finalize(void* wsbase, float* __restrict__ out) {
  Ws ws = makeWs(wsbase);
  int sel[MAX_DET];
  int cnt = 0;
  for (int i = 0; i < K_PRE && cnt < MAX_DET; ++i)
    if ((ws.keepw[i >> 5] >> (i & 31)) & 1u) sel[cnt++] = i;
  for (int i = 0; i < K_PRE && cnt < MAX_DET; ++i)
    if (!((ws.keepw[i >> 5] >> (i & 31)) & 1u)) sel[cnt++] = i;
  for (int s = 0; s < MAX_DET; ++s) {
    int i = sel[s];
    bool v = ((ws.keepw[i >> 5] >> (i & 31)) & 1u) != 0;
    for (int d = 0; d < 7; ++d)
      out[s * 8 + d] = v ? ws.boxes[i * 7 + d] : 0.0f;
    out[s * 8 + 7] = v ? ws.score[i] : 0.0f;
    out[800 + s] = (float)ws.label[i];   // labels NOT masked (matches reference)
    out[900 + s] = v ? 1.0f : 0.0f;
  }
}

// ---------------- launcher ----------------
extern "C" void kernel_launch(void* const* d_in, const int* in_sizes, int n_in,
                              void* d_out, int out_size, void* d_ws, size_t ws_size,
                              hipStream_t stream) {
  (void)in_sizes; (void)n_in; (void)out_size; (void)ws_size;
  const float* cls = (const float*)d_in[0];   // (4,10,512,512) fp32
  const float* bbx = (const float*)d_in[1];   // (4,7,512,512) fp32
  float* out = (float*)d_out;

  // zero hist1 + hist2 + scalars + keep words
  int nz = 65536 + 65536 + 64 + 64;
  dpp_zero<<<(nz + 255) / 256, 256, 0, stream>>>((uint32*)d_ws, nz);

  dpp_hist_hi<<<1280, 256, 0, stream>>>(cls, d_ws);
  dpp_scan<<<1, 1024, 0, stream>>>(d_ws, 0);
  dpp_hist_lo<<<2560, 256, 0, stream>>>(cls, d_ws);
  dpp_scan<<<1, 1024, 0, stream>>>(d_ws, 1);
  dpp_compact<<<2560, 256, 0, stream>>>(cls, d_ws);
  dpp_sort<<<1, 1024, 0, stream>>>(d_ws);
  dpp_decode<<<K_PRE / 256, 256, 0, stream>>>(bbx, d_ws);
  dpp_suppress<<<2048, 256, 0, stream>>>(d_ws);   // 16384 tiles, 8 waves/block
  dpp_nms<<<1, 32, 0, stream>>>(d_ws);
  dpp_finalize<<<1, 1, 0, stream>>>(d_ws, out);
}